// Decoder_77841987272826
// MI455X (gfx1250) — compile-verified
//
#include <hip/hip_runtime.h>

typedef __attribute__((ext_vector_type(2))) float v2f;
typedef __attribute__((ext_vector_type(8))) float v8f;

#define NL    128
#define BATCH 256
#define H     64
#define C     64
// K-pair interleaved LDS layout: 32 kp-rows, stride 160 floats (160%64==32 ->
// lanes 0-15 (kp) and 16-31 (kp+1) hit disjoint bank halves; conflict-free b64)
#define WSTRIDE 160

// ---------------------------------------------------------------------------
// Kernel 1: h = BN(ReLU(latent @ W1 + b1))  -> hbn [256 x 64] in workspace
// ---------------------------------------------------------------------------
__global__ __launch_bounds__(64) void mlp_bn_kernel(
    const float* __restrict__ latent, const float* __restrict__ W1,
    const float* __restrict__ b1, const float* __restrict__ gamma,
    const float* __restrict__ beta, const float* __restrict__ mean,
    const float* __restrict__ var, float* __restrict__ hbn) {
  const int b = blockIdx.x;    // 0..255
  const int j = threadIdx.x;   // 0..63 (output column)
  const float* lrow = latent + b * NL;
  float acc = b1[j];
#pragma unroll 4
  for (int k = 0; k < NL; ++k) acc = fmaf(lrow[k], W1[k * H + j], acc);
  acc = fmaxf(acc, 0.0f);
  hbn[b * H + j] = (acc - mean[j]) * rsqrtf(var[j] + 1e-5f) * gamma[j] + beta[j];
}

// ---------------------------------------------------------------------------
// Kernel 2: out[b,g,c] = sum_h hbn[b,h] * weight_table[genes_oi[g],h,c] + bias
// One block per gene. 512 threads = 16 waves; wave w computes rows
// [16w, 16w+16) x all 64 columns via 4 accumulator tiles of 16x16 using
// v_wmma_f32_16x16x4_f32 (K-loop of 16).
// ---------------------------------------------------------------------------
__global__ __launch_bounds__(512) void decoder_wmma_kernel(
    const float* __restrict__ hbn, const int* __restrict__ genes_oi,
    const float* __restrict__ weight_table, const float* __restrict__ bias_table,
    float* __restrict__ out, int G) {
  // s_w[kp][c][2] = { w[2kp][c], w[2kp+1][c] },  padded row stride WSTRIDE
  __shared__ float s_w[(H / 2) * WSTRIDE];  // 20 KB

  const int g    = blockIdx.x;
  const int gene = genes_oi[g];
  const int tid  = threadIdx.x;

  // Cooperative stage: 64x64 row-major tile -> K-pair interleaved LDS layout.
  {
    const float4* src = (const float4*)(weight_table + (size_t)gene * (H * C));
#pragma unroll
    for (int it = 0; it < (H * C / 4) / 512; ++it) {  // 1024 float4 / 512 thr
      const int f = tid + it * 512;     // float4 index
      const float4 v = src[f];
      const int h = f >> 4;             // source row 0..63
      const int c = (f & 15) << 2;      // source col 0..60
      float* dst = s_w + (h >> 1) * WSTRIDE + (c << 1) + (h & 1);
      dst[0] = v.x; dst[2] = v.y; dst[4] = v.z; dst[6] = v.w;
    }
  }
  __syncthreads();

  const int wave = tid >> 5;
  const int lane = tid & 31;
  const int lo   = lane & 15;   // N / M position within half-wave
  const int hi   = lane >> 4;   // half-wave select
  const int m0   = wave * 16;   // this wave's row tile

  v8f acc0 = {}, acc1 = {}, acc2 = {}, acc3 = {};

  // A-matrix 16x4 f32 layout: lanes 0-15 hold K={k0,k0+1}, lanes 16-31 K={k0+2,k0+3}
  const float* arow = hbn + (m0 + lo) * H + 2 * hi;

#pragma unroll
  for (int k0 = 0; k0 < H; k0 += 4) {
    v2f a = *(const v2f*)(arow + k0);   // 8B-aligned global load (L2-hot)

    // B-matrix 4x16 f32 fragment: one aligned ds_load_b64 per n-tile.
    // kp = kk/2 where kk = k0 + 2*hi;  (k0>>1) is even, so kp = k0/2 + hi.
    const float* bbase = s_w + ((k0 >> 1) + hi) * WSTRIDE + (lo << 1);
    v2f b0  = *(const v2f*)(bbase + 0);    // cols  0..15
    v2f b1v = *(const v2f*)(bbase + 32);   // cols 16..31
    v2f b2  = *(const v2f*)(bbase + 64);   // cols 32..47
    v2f b3  = *(const v2f*)(bbase + 96);   // cols 48..63

    acc0 = __builtin_amdgcn_wmma_f32_16x16x4_f32(false, a, false, b0,  (short)0, acc0, false, false);
    acc1 = __builtin_amdgcn_wmma_f32_16x16x4_f32(false, a, false, b1v, (short)0, acc1, false, false);
    acc2 = __builtin_amdgcn_wmma_f32_16x16x4_f32(false, a, false, b2,  (short)0, acc2, false, false);
    acc3 = __builtin_amdgcn_wmma_f32_16x16x4_f32(false, a, false, b3,  (short)0, acc3, false, false);
  }

  // Epilogue: C/D layout — VGPR j: lanes 0-15 = (M=j, N=lane), lanes 16-31 = (M=j+8).
  const float* bias = bias_table + (size_t)gene * C;
  const float bs0 = bias[lo], bs1 = bias[16 + lo], bs2 = bias[32 + lo], bs3 = bias[48 + lo];
#pragma unroll
  for (int j = 0; j < 8; ++j) {
    const int row = m0 + j + 8 * hi;
    float* orow = out + ((size_t)row * G + g) * C + lo;
    orow[0]  = acc0[j] + bs0;
    orow[16] = acc1[j] + bs1;
    orow[32] = acc2[j] + bs2;
    orow[48] = acc3[j] + bs3;
  }
}

// ---------------------------------------------------------------------------
extern "C" void kernel_launch(void* const* d_in, const int* in_sizes, int n_in,
                              void* d_out, int out_size, void* d_ws, size_t ws_size,
                              hipStream_t stream) {
  const float* latent   = (const float*)d_in[0];
  const int*   genes_oi = (const int*)d_in[1];
  const float* W1       = (const float*)d_in[2];
  const float* b1       = (const float*)d_in[3];
  const float* gamma    = (const float*)d_in[4];
  const float* beta     = (const float*)d_in[5];
  const float* mean     = (const float*)d_in[6];
  const float* var      = (const float*)d_in[7];
  const float* wt       = (const float*)d_in[8];
  const float* bt       = (const float*)d_in[9];
  float* out = (float*)d_out;
  float* hbn = (float*)d_ws;          // 256*64 floats = 64 KB scratch
  const int G = in_sizes[1];          // number of gathered genes (5000)

  mlp_bn_kernel<<<BATCH, H, 0, stream>>>(latent, W1, b1, gamma, beta, mean, var, hbn);
  decoder_wmma_kernel<<<G, 512, 0, stream>>>(hbn, genes_oi, wt, bt, out, G);
}